// LTFGW_52175262712150
// MI455X (gfx1250) — compile-verified
//
#include <hip/hip_runtime.h>
#include <hip/hip_bf16.h>

typedef __attribute__((ext_vector_type(2))) float v2f;
typedef __attribute__((ext_vector_type(8))) float v8f;

#define NN    15000
#define FD    16
#define TT    16
#define MM    10
#define KK    17
#define NCG   5
#define NSK   25
#define EPSR  0.05f
#define WPB   8            // waves per block (nodes per block)
#define TSTR  12           // LDS tile row stride (floats): 48B, 16B aligned

__device__ __forceinline__ float wave_sum(float v) {
#pragma unroll
  for (int m = 16; m >= 1; m >>= 1) v += __shfl_xor(v, m, 32);
  return v;
}
__device__ __forceinline__ float wave_max(float v) {
#pragma unroll
  for (int m = 16; m >= 1; m >>= 1) v = fmaxf(v, __shfl_xor(v, m, 32));
  return v;
}

__global__ __launch_bounds__(256)
void ltfgw_fgw_kernel(const float* __restrict__ x,
                      const int*   __restrict__ neighbors,
                      const float* __restrict__ templates,
                      const float* __restrict__ tfeat,
                      const float* __restrict__ alpha,
                      float*       __restrict__ out) {
  __shared__ float ldsC2[MM * MM];        // template adjacency (10x10)
  __shared__ float ldsCC[MM];             // (C2*C2) @ h2 per column m
  __shared__ float ldsN2[MM];             // ||F2_m||^2
  __shared__ float tile[WPB][KK * TSTR];  // per-wave Mcost-dot transpose tile

  const int tid  = threadIdx.x;
  const int wave = tid >> 5;
  const int lane = tid & 31;
  const int t    = blockIdx.x % TT;                 // template id (uniform per block)
  const int n    = (blockIdx.x / TT) * WPB + wave;  // node id (one per wave)

  const float a = alpha[0];

  // ---- block-shared per-template constants ----
  if (tid < MM * MM) ldsC2[tid] = templates[t * MM * MM + tid];
  if (tid < MM) {
    float s = 0.f, s2 = 0.f;
#pragma unroll
    for (int j = 0; j < MM; ++j) { float c = templates[t * MM * MM + tid * MM + j]; s += c * c; }
#pragma unroll
    for (int f = 0; f < FD; ++f) { float v = tfeat[t * MM * FD + tid * FD + f]; s2 += v * v; }
    ldsCC[tid] = s * (1.0f / (float)MM);
    ldsN2[tid] = s2;
  }

  // ---- WMMA: dot[k][m] = <F1_k, F2_m>, rows k=0..15 via f32 16x16x4, 4 K-chunks ----
  const int  arow  = lane & 15;            // A row (M)
  const int  kp    = (lane >> 4) << 1;     // K sub-pair: 0 or 2
  const int  col   = lane & 15;            // B/D column (N)
  const int  colC  = (col < MM) ? col : 0; // clamp: keep loads in-bounds
  const long baseA = (long)neighbors[n * KK + arow] * FD;
  const long baseB = (long)t * MM * FD + (long)colC * FD;

  v8f acc = {0.f, 0.f, 0.f, 0.f, 0.f, 0.f, 0.f, 0.f};
#pragma unroll
  for (int c = 0; c < 4; ++c) {
    const int f0 = 4 * c;
    v2f av, bv;
    av.x = x[baseA + f0 + kp];
    av.y = x[baseA + f0 + kp + 1];
    bv.x = tfeat[baseB + f0 + kp];
    bv.y = tfeat[baseB + f0 + kp + 1];
    acc = __builtin_amdgcn_wmma_f32_16x16x4_f32(false, av, false, bv,
                                                (short)0, acc, false, false);
  }
  // scatter D (column-per-lane layout) into LDS as row-major [k][m]
  if (col < MM) {
    const int mhi = (lane >> 4) << 3;  // +8 rows for upper half-wave
#pragma unroll
    for (int r = 0; r < 8; ++r) tile[wave][(r + mhi) * TSTR + col] = acc[r];
  }
  // row k=16 by direct dot products (lanes 0..9, one column each)
  if (lane < MM) {
    const long b16 = (long)neighbors[n * KK + (KK - 1)] * FD;
    const long bf  = (long)t * MM * FD + (long)lane * FD;
    float d = 0.f;
#pragma unroll
    for (int f = 0; f < FD; ++f) d += x[b16 + f] * tfeat[bf + f];
    tile[wave][(KK - 1) * TSTR + lane] = d;
  }

  // per-lane ego-row squared norm ||F1_k||^2 (lane k owns row k; lanes>=17 alias row 16)
  const int  kkr = (lane < KK) ? lane : (KK - 1);
  const bool act = lane < KK;
  const long baseR = (long)neighbors[n * KK + kkr] * FD;
  float n1 = 0.f;
#pragma unroll
  for (int f = 0; f < FD; ++f) { float v = x[baseR + f]; n1 += v * v; }

  __syncthreads();

  // ---- build per-lane rows: Mcost, constC; init product coupling ----
  float Mc[MM], cC[MM], Tc[MM], lK[MM], gm[MM];
  const float c1k = (kkr == 0) ? (16.0f / 17.0f) : (1.0f / 17.0f);  // (C1.^2)@h1
#pragma unroll
  for (int m = 0; m < MM; ++m) {
    float dot = tile[wave][kkr * TSTR + m];
    Mc[m] = n1 + ldsN2[m] - 2.0f * dot;
    cC[m] = c1k + ldsCC[m];
    Tc[m] = 1.0f / (float)(KK * MM);
  }

  const float logh1 = -2.83321334f;  // log(1/17)
  const float logh2 = -2.30258509f;  // log(1/10)
  const float NEGBIG = -3.4e38f;
  float phi = 0.f;

  // ---- conditional-gradient loop ----
#pragma unroll 1
  for (int itcg = 0; itcg < NCG; ++itcg) {
    // u = sum_{k>=1} Tc[k,:]  (butterfly), v = Tc[0,:] (broadcast); lane 0 uses u.
    float uv[MM];
#pragma unroll
    for (int m = 0; m < MM; ++m) {
      float contrib = (lane >= 1 && lane < KK) ? Tc[m] : 0.f;
      float u = wave_sum(contrib);
      float v = __shfl(Tc[m], 0, 32);
      uv[m] = (kkr == 0) ? u : v;
    }
    // G row = (1-a)*Mcost + 2a*(constC - 2*(C1 Tc C2^T)); track row abs-max
    float rmax = 0.f;
#pragma unroll
    for (int m = 0; m < MM; ++m) {
      float r = 0.f;
#pragma unroll
      for (int j = 0; j < MM; ++j) r += uv[j] * ldsC2[m * MM + j];
      float tens = cC[m] - 2.0f * r;
      float g = (1.0f - a) * Mc[m] + (2.0f * a) * tens;
      lK[m] = g;
      rmax = fmaxf(rmax, fabsf(g));
    }
    float scale = wave_max(act ? rmax : 0.f) + 1e-12f;
    float sc = -1.0f / (scale * EPSR);
#pragma unroll
    for (int m = 0; m < MM; ++m) lK[m] *= sc;   // logKmat = -(G/scale)/eps

    // ---- log-domain Sinkhorn (phi = f/eps, gm = g/eps) ----
    phi = 0.f;
#pragma unroll
    for (int m = 0; m < MM; ++m) gm[m] = 0.f;
#pragma unroll 1
    for (int si = 0; si < NSK; ++si) {
      // f-update: per-lane logsumexp over m
      float mx = NEGBIG;
#pragma unroll
      for (int m = 0; m < MM; ++m) mx = fmaxf(mx, lK[m] + gm[m]);
      float s = 0.f;
#pragma unroll
      for (int m = 0; m < MM; ++m) s += __expf(lK[m] + gm[m] - mx);
      phi = logh1 - (mx + __logf(s));
      // g-update: cross-lane logsumexp over k for each column m
#pragma unroll
      for (int m = 0; m < MM; ++m) {
        float tv = lK[m] + phi;
        float cm = wave_max(act ? tv : NEGBIG);
        float e  = act ? __expf(tv - cm) : 0.f;
        float cs = wave_sum(e);
        gm[m] = logh2 - (cm + __logf(cs));
      }
    }
    // CG step: Tc += 2/(it+2) * (S - Tc), S = exp(logK + phi + gm)
    float step = 2.0f / ((float)itcg + 2.0f);
#pragma unroll
    for (int m = 0; m < MM; ++m) {
      float S = __expf(lK[m] + phi + gm[m]);
      Tc[m] += step * (S - Tc[m]);
    }
  }

  // ---- final FGW value: (1-a)*sum(Mcost.*Tc) + a*sum(tens(Tc).*Tc) ----
  float uv[MM];
#pragma unroll
  for (int m = 0; m < MM; ++m) {
    float contrib = (lane >= 1 && lane < KK) ? Tc[m] : 0.f;
    float u = wave_sum(contrib);
    float v = __shfl(Tc[m], 0, 32);
    uv[m] = (kkr == 0) ? u : v;
  }
  float part = 0.f;
#pragma unroll
  for (int m = 0; m < MM; ++m) {
    float r = 0.f;
#pragma unroll
    for (int j = 0; j < MM; ++j) r += uv[j] * ldsC2[m * MM + j];
    float tens = cC[m] - 2.0f * r;
    part += ((1.0f - a) * Mc[m] + a * tens) * Tc[m];
  }
  float fgw = wave_sum(act ? part : 0.f);
  if (lane == 0) out[(long)n * TT + t] = fgw;
}

extern "C" void kernel_launch(void* const* d_in, const int* in_sizes, int n_in,
                              void* d_out, int out_size, void* d_ws, size_t ws_size,
                              hipStream_t stream) {
  (void)in_sizes; (void)n_in; (void)out_size; (void)d_ws; (void)ws_size;
  const float* x         = (const float*)d_in[0];
  // d_in[1] = edge_index (raw COO) — unused; `neighbors` is its padded form
  const int*   neighbors = (const int*)  d_in[2];
  const float* templates = (const float*)d_in[3];
  const float* tfeat     = (const float*)d_in[4];
  const float* alpha     = (const float*)d_in[5];
  float*       out       = (float*)d_out;

  const int blocks = TT * ((NN + WPB - 1) / WPB);   // 16 * 1875 = 30000
  ltfgw_fgw_kernel<<<blocks, 256, 0, stream>>>(x, neighbors, templates, tfeat, alpha, out);
}